// LSTM_24970939859050
// MI455X (gfx1250) — compile-verified
//
#include <hip/hip_runtime.h>
#include <hip/hip_bf16.h>

// V=100000, D=300, H=16, C=5, B=256, T=128
// Phase 1: xg[b*T+t, g] = sum_d emb[x[b,t],d]*W_ih[g,d] + b_ih[g] + b_hh[g]
//   M = B*T = 32768 rows, K = 300, N = 4H = 64 -> fp32 WMMA 16x16x4 tiles.
// Phase 2: per-batch-independent LSTM scan (16 waves, 16 batch rows each),
//   recurrent GEMM h@W_hh^T and output GEMM h@W_out^T both via WMMA.

typedef __attribute__((ext_vector_type(2))) float v2f;
typedef __attribute__((ext_vector_type(8))) float v8f;

#define WMMA_F32X4(a, b, c)                                                  \
    __builtin_amdgcn_wmma_f32_16x16x4_f32(false, (a), false, (b), (short)0,  \
                                          (c), false, false)

constexpr int kD = 300;   // embedding dim (K of input GEMM)
constexpr int kH = 16;    // hidden
constexpr int kG = 64;    // 4*H gates
constexpr int kB = 256;   // batch
constexpr int kT = 128;   // time
constexpr int kC = 5;     // classes

__device__ __forceinline__ float sigf(float v) {
    return 1.0f / (1.0f + __expf(-v));
}

// -------- Phase 1: fused embedding gather + input GEMM ---------------------
// grid: 256 blocks x 256 threads (8 waves). Each wave owns one 16-row M-tile.
__global__ __launch_bounds__(256) void lstm_xgemm(
    const int* __restrict__ x, const float* __restrict__ emb,
    const float* __restrict__ W_ih, const float* __restrict__ b_ih,
    const float* __restrict__ b_hh, float* __restrict__ xg) {
    const int lane = threadIdx.x & 31;
    const int wave = threadIdx.x >> 5;
    const int tile = blockIdx.x * 8 + wave;     // 2048 tiles total
    const int rowbase = tile * 16;
    const int m    = lane & 15;   // A row / C-D column index
    const int half = lane >> 4;   // K-split for A/B, M+8 split for C/D
    const int kb   = 2 * half;
    const int n    = m;           // B/C/D column index (gate within tile)

    // One token row per A-matrix row (lanes 16-31 reload same rows: cheap).
    const int tok = x[rowbase + m];
    const float* __restrict__ arow = emb + (size_t)tok * kD;

    // Accumulators for 4 N-tiles of 16 gates; bias folded in.
    v8f acc[4];
#pragma unroll
    for (int j = 0; j < 4; ++j) {
        const float bias = b_ih[j * 16 + n] + b_hh[j * 16 + n];
#pragma unroll
        for (int r = 0; r < 8; ++r) acc[j][r] = bias;
    }

    // K loop: 300 = 75 chunks of 4. A loaded once per chunk, reused 4x.
    for (int k = 0; k < kD; k += 4) {
        v2f a;
        a.x = arow[k + kb];
        a.y = arow[k + kb + 1];
#pragma unroll
        for (int j = 0; j < 4; ++j) {
            const float* __restrict__ wr =
                W_ih + (size_t)(j * 16 + n) * kD + k + kb;  // B[k'][g]=W_ih[g][k']
            v2f b;
            b.x = wr[0];
            b.y = wr[1];
            acc[j] = WMMA_F32X4(a, b, acc[j]);
        }
    }

    // Store xg row-major [row][g]; D-layout: row = rowbase + r + 8*half.
#pragma unroll
    for (int j = 0; j < 4; ++j)
#pragma unroll
        for (int r = 0; r < 8; ++r) {
            const int row = rowbase + r + 8 * half;
            xg[(size_t)row * kG + j * 16 + n] = acc[j][r];
        }
}

// -------- Phase 2: LSTM scan + output projection ---------------------------
// grid: 16 blocks x 32 threads. Block bi owns batch rows [16*bi, 16*bi+16).
__global__ __launch_bounds__(32) void lstm_scan(
    const float* __restrict__ xg, const float* __restrict__ W_hh,
    const float* __restrict__ W_out, const float* __restrict__ b_out,
    float* __restrict__ out) {
    __shared__ float lds_h[16 * 16];  // h tile staging (D-layout -> A-layout)

    const int lane = threadIdx.x & 31;
    const int m    = lane & 15;
    const int half = lane >> 4;
    const int kb   = 2 * half;
    const int n    = m;
    const int b0   = blockIdx.x * 16;

    // Preload B operands into registers (tiny, L2-resident).
    // Recurrent: B[k][g] = W_hh[g][k], K=16 -> 4 chunks, 4 N-tiles.
    v2f bh[4][4];
#pragma unroll
    for (int j = 0; j < 4; ++j)
#pragma unroll
        for (int kk = 0; kk < 4; ++kk) {
            const float* __restrict__ wr =
                W_hh + (size_t)(j * 16 + n) * kH + kk * 4 + kb;
            bh[j][kk].x = wr[0];
            bh[j][kk].y = wr[1];
        }
    // Output: B[k][c] = W_out[c][k], zero-padded for c in [5,16).
    v2f bo[4];
#pragma unroll
    for (int kk = 0; kk < 4; ++kk) {
        if (n < kC) {
            const float* __restrict__ wr = W_out + (size_t)n * kH + kk * 4 + kb;
            bo[kk].x = wr[0];
            bo[kk].y = wr[1];
        } else {
            bo[kk].x = 0.0f;
            bo[kk].y = 0.0f;
        }
    }
    const float biaso = (n < kC) ? b_out[n] : 0.0f;

    // State: h (A-layout fragments) and c, start at zero.
    v2f a[4];
#pragma unroll
    for (int kk = 0; kk < 4; ++kk) { a[kk].x = 0.0f; a[kk].y = 0.0f; }
    float cst[8];
#pragma unroll
    for (int r = 0; r < 8; ++r) cst[r] = 0.0f;

    for (int t = 0; t < kT; ++t) {
        // Gates initialized from precomputed xg (bias already folded in).
        v8f g[4];
#pragma unroll
        for (int j = 0; j < 4; ++j)
#pragma unroll
            for (int r = 0; r < 8; ++r) {
                const size_t row = (size_t)(b0 + r + 8 * half) * kT + t;
                g[j][r] = xg[row * kG + j * 16 + n];
            }

        // g += h_{t-1} @ W_hh^T : 4 K-chunks x 4 gate tiles of WMMA.
#pragma unroll
        for (int kk = 0; kk < 4; ++kk)
#pragma unroll
            for (int j = 0; j < 4; ++j) g[j] = WMMA_F32X4(a[kk], bh[j][kk], g[j]);

        // Elementwise LSTM cell (PyTorch gate order i,f,g,o).
        float hval[8];
#pragma unroll
        for (int r = 0; r < 8; ++r) {
            const float ig = sigf(g[0][r]);
            const float fg = sigf(g[1][r]);
            const float gg = tanhf(g[2][r]);
            const float og = sigf(g[3][r]);
            cst[r]  = fg * cst[r] + ig * gg;
            hval[r] = og * tanhf(cst[r]);
        }

        // D-layout -> A-layout transpose for h via LDS.
#pragma unroll
        for (int r = 0; r < 8; ++r) lds_h[(r + 8 * half) * 16 + n] = hval[r];
        __syncthreads();
#pragma unroll
        for (int kk = 0; kk < 4; ++kk) {
            a[kk].x = lds_h[m * 16 + kk * 4 + kb];
            a[kk].y = lds_h[m * 16 + kk * 4 + kb + 1];
        }
        __syncthreads();

        // Output projection: out[b,t,:] = h_t @ W_out^T + b_out (WMMA, padded).
        v8f oacc;
#pragma unroll
        for (int r = 0; r < 8; ++r) oacc[r] = biaso;
#pragma unroll
        for (int kk = 0; kk < 4; ++kk) oacc = WMMA_F32X4(a[kk], bo[kk], oacc);

        if (n < kC) {
#pragma unroll
            for (int r = 0; r < 8; ++r) {
                const size_t row = (size_t)(b0 + r + 8 * half) * kT + t;
                out[row * kC + n] = oacc[r];
            }
        }
    }
}

extern "C" void kernel_launch(void* const* d_in, const int* in_sizes, int n_in,
                              void* d_out, int out_size, void* d_ws,
                              size_t ws_size, hipStream_t stream) {
    const int*   x     = (const int*)d_in[0];
    const float* emb   = (const float*)d_in[1];
    const float* W_ih  = (const float*)d_in[2];
    const float* W_hh  = (const float*)d_in[3];
    const float* b_ih  = (const float*)d_in[4];
    const float* b_hh  = (const float*)d_in[5];
    const float* W_out = (const float*)d_in[6];
    const float* b_out = (const float*)d_in[7];

    float* xg  = (float*)d_ws;   // [B*T, 64] f32 = 8.39 MB
    float* out = (float*)d_out;  // [B, T, C] f32

    // 2048 M-tiles / 8 waves per block.
    lstm_xgemm<<<256, 256, 0, stream>>>(x, emb, W_ih, b_ih, b_hh, xg);
    // 16 batch tiles, one wave each; sequential scan over T=128 inside.
    lstm_scan<<<16, 32, 0, stream>>>(xg, W_hh, W_out, b_out, out);
}